// LinearAE_CNN_Mamba_83004537963173
// MI455X (gfx1250) — compile-verified
//
#include <hip/hip_runtime.h>
#include <hip/hip_bf16.h>

// ---------------------------------------------------------------------------
// Model dims (from reference)
// ---------------------------------------------------------------------------
#define BB     32
#define SS     512
#define EE     128
#define HH     1024
#define COUT   256
#define DM     256          // D_MODEL
#define DI     512          // D_INNER
#define DS     16           // D_STATE
#define DTR    16           // DT_RANK
#define TT     256          // mamba sequence length (S/2)
#define NTOK   (BB*TT)      // 8192 tokens
#define KS_FC1 8            // grid-level K split for encoder FC1

typedef float v2f __attribute__((ext_vector_type(2)));
typedef float v8f __attribute__((ext_vector_type(8)));

// ---------------------------------------------------------------------------
// Generic fp32 WMMA GEMM:  C(M,N) = act(A(M,K) * B(K,N) + bias)
//   - all strides / K compile-time -> incremental pointer arithmetic and a
//     *uniform* compile-time trip count (scalar loop control, unrollable)
//   - block = 256 threads (8 waves); waves split K in chunks of 4 (WMMA K=4)
//   - each block computes a 32-row x 16-col tile (2 M-tiles reuse B loads)
//   - KS > 1: grid.z splits K; raw partial sums go to per-slice buffers
//     (deterministic fixed-order reduction in a separate epilogue kernel)
//   ACT: 0 = none, 1 = relu, 2 = softplus
// ---------------------------------------------------------------------------
template <int ACT, bool HAS_BIAS, int LDA, int LDB, int LDC, int K, int KS>
__global__ void wmma_gemm_f32(const float* __restrict__ A,
                              const float* __restrict__ Bm,
                              float* __restrict__ C,
                              const float* __restrict__ bias)
{
    const int nt   = blockIdx.x * 16;       // N tile start
    const int mt0  = blockIdx.y * 32;       // M block start (2 tiles of 16)
    const int wave = threadIdx.x >> 5;      // 0..7
    const int lane = threadIdx.x & 31;
    const int half = lane >> 4;             // 0/1
    const int l16  = lane & 15;

    if (KS > 1) {                           // K-split: shift A cols / B rows,
        const int kz = blockIdx.z;          // write raw partials per slice
        A  += (size_t)kz * K;
        Bm += (size_t)kz * K * LDB;
        C  += (size_t)kz * (size_t)gridDim.y * 32 * LDC;
    }

    v8f acc0 = {0.f,0.f,0.f,0.f,0.f,0.f,0.f,0.f};
    v8f acc1 = {0.f,0.f,0.f,0.f,0.f,0.f,0.f,0.f};

    // lane-resident stream pointers (constant strides)
    const float* ap0 = A + (size_t)(mt0 + l16) * LDA + wave * 4 + 2 * half;
    const float* ap1 = ap0 + (size_t)16 * LDA;
    const float* bp  = Bm + nt + l16 + (size_t)(wave * 4 + 2 * half) * LDB;

    // uniform trip count: scalar loop control + real unrolling
    constexpr int TRIPS = (K + 31) / 32;
    if ((K % 32 == 0) || (wave * 4 < K)) {
        #pragma unroll 4
        for (int it = 0; it < TRIPS; ++it) {
            v2f a0, a1, b;
            a0.x = ap0[0];  a0.y = ap0[1];
            a1.x = ap1[0];  a1.y = ap1[1];
            b.x  = bp[0];   b.y  = bp[LDB];
            if (K >= 1024)  // big streamed weights only; speculative, OOB-safe
                __builtin_prefetch(bp + (size_t)256 * LDB, 0, 1);
            acc0 = __builtin_amdgcn_wmma_f32_16x16x4_f32(false, a0, false, b,
                                                         (short)0, acc0,
                                                         false, false);
            acc1 = __builtin_amdgcn_wmma_f32_16x16x4_f32(false, a1, false, b,
                                                         (short)0, acc1,
                                                         false, false);
            ap0 += 32;  ap1 += 32;  bp += (size_t)32 * LDB;
        }
    }

    // cross-wave reduction: 8 waves x 2 tiles x 8 vgprs x 32 lanes = 16 KB LDS
    __shared__ float red[8][2][8][32];
    #pragma unroll
    for (int i = 0; i < 8; i++) {
        red[wave][0][i][lane] = acc0[i];
        red[wave][1][i][lane] = acc1[i];
    }
    __syncthreads();

    if (wave == 0) {
        #pragma unroll
        for (int t = 0; t < 2; t++) {
            #pragma unroll
            for (int i = 0; i < 8; i++) {
                float s = 0.f;
                #pragma unroll
                for (int w = 0; w < 8; w++) s += red[w][t][i][lane];
                const int row = mt0 + t * 16 + i + 8 * half;  // C layout: vgpr i,
                const int col = nt + l16;                     // lane-half = +8 rows
                if (KS == 1) {
                    if (HAS_BIAS) s += bias[col];
                    if (ACT == 1) s = fmaxf(s, 0.f);
                    if (ACT == 2) s = (s > 20.f) ? s : log1pf(__expf(s));
                }
                C[(size_t)row * LDC + col] = s;
            }
        }
    }
}

// ---------------------------------------------------------------------------
// fixed-order K-slice reduction + bias + relu for FC1 (deterministic)
// ---------------------------------------------------------------------------
__global__ void ksum_bias_relu_kernel(const float* __restrict__ part,
                                      const float* __restrict__ bias,
                                      float* __restrict__ C)
{
    const int i = blockIdx.x * blockDim.x + threadIdx.x;
    if (i >= BB * HH) return;
    float s = 0.f;
    #pragma unroll
    for (int z = 0; z < KS_FC1; z++) s += part[z * BB * HH + i];
    s += bias[i & (HH - 1)];
    C[i] = fmaxf(s, 0.f);
}

// ---------------------------------------------------------------------------
// Embedding gather: h0[b*S + s][e] = emb[x[b*S+s]][e]
// ---------------------------------------------------------------------------
__global__ void embed_kernel(const int* __restrict__ x,
                             const float* __restrict__ emb,
                             float* __restrict__ h0)
{
    const size_t i = (size_t)blockIdx.x * blockDim.x + threadIdx.x;
    if (i >= (size_t)BB * SS * EE) return;
    const int e     = (int)(i & (EE - 1));
    const size_t rw = i >> 7;                 // b*S + s
    h0[i] = emb[(size_t)x[rw] * EE + e];
}

// ---------------------------------------------------------------------------
// conv weight transpose: (oc,e,k) -> (e*5+k, oc) for coalesced reads
// ---------------------------------------------------------------------------
__global__ void convw_transpose_kernel(const float* __restrict__ w,
                                       float* __restrict__ wT)
{
    const int i = blockIdx.x * blockDim.x + threadIdx.x;
    if (i >= COUT * EE * 5) return;
    const int oc = i / (EE * 5);
    const int r  = i - oc * (EE * 5);         // e*5 + k
    wT[r * COUT + oc] = w[i];
}

// ---------------------------------------------------------------------------
// conv1d(E=128 -> 256, k=5, pad 2) + bias + relu + maxpool2 over time.
// One block per (b, pooled-pos), 256 threads = out channels.
// ---------------------------------------------------------------------------
__global__ void conv_pool_kernel(const float* __restrict__ h2,
                                 const float* __restrict__ wT,
                                 const float* __restrict__ cb,
                                 float* __restrict__ u)
{
    const int sp = blockIdx.x & (TT - 1);
    const int b  = blockIdx.x >> 8;
    const int oc = threadIdx.x;

    __shared__ float win[6 * EE];             // positions 2sp-2 .. 2sp+3
    for (int j = threadIdx.x; j < 6 * EE; j += 256) {
        const int p = j >> 7, e = j & (EE - 1);
        const int s = 2 * sp - 2 + p;
        win[j] = (s >= 0 && s < SS) ? h2[((size_t)(b * SS + s)) * EE + e] : 0.f;
    }
    __syncthreads();

    float a0 = 0.f, a1 = 0.f;
    for (int e = 0; e < EE; e++) {
        #pragma unroll
        for (int k = 0; k < 5; k++) {
            const float w = wT[(e * 5 + k) * COUT + oc];
            a0 = fmaf(win[k * EE + e],       w, a0);
            a1 = fmaf(win[(k + 1) * EE + e], w, a1);
        }
    }
    a0 = fmaxf(a0 + cb[oc], 0.f);
    a1 = fmaxf(a1 + cb[oc], 0.f);
    u[(size_t)blockIdx.x * COUT + oc] = fmaxf(a0, a1);
}

// ---------------------------------------------------------------------------
// depthwise causal conv (k=4, pad 3) over time + SiLU.  xm = xz[:, :512]
// ---------------------------------------------------------------------------
__global__ void dwconv_silu_kernel(const float* __restrict__ xz,
                                   const float* __restrict__ mw,
                                   const float* __restrict__ mb,
                                   float* __restrict__ xc)
{
    const size_t i = (size_t)blockIdx.x * blockDim.x + threadIdx.x;
    if (i >= (size_t)NTOK * DI) return;
    const int d   = (int)(i & (DI - 1));
    const int tok = (int)(i >> 9);
    const int t   = tok & (TT - 1);
    const int b   = tok >> 8;

    const float* base = xz + (size_t)(b * TT) * (2 * DI) + d;
    float acc = mb[d];
    if (t >= 3) acc = fmaf(base[(size_t)(t - 3) * (2 * DI)], mw[d * 4 + 0], acc);
    if (t >= 2) acc = fmaf(base[(size_t)(t - 2) * (2 * DI)], mw[d * 4 + 1], acc);
    if (t >= 1) acc = fmaf(base[(size_t)(t - 1) * (2 * DI)], mw[d * 4 + 2], acc);
    acc = fmaf(base[(size_t)t * (2 * DI)], mw[d * 4 + 3], acc);
    const float sig = 1.f / (1.f + __expf(-acc));
    xc[i] = acc * sig;
}

// ---------------------------------------------------------------------------
// Selective scan: 32 blocks (batch) x 512 threads (channel d).
// 16 states live in registers; per-step B/C broadcast via LDS.
// Writes y_inner = (scan_y + xc*D) * silu(z).
// ---------------------------------------------------------------------------
__global__ void __launch_bounds__(DI)
mamba_scan_kernel(const float* __restrict__ xc,
                  const float* __restrict__ dtb,
                  const float* __restrict__ dbc,
                  const float* __restrict__ xz,
                  const float* __restrict__ A_log,
                  const float* __restrict__ Dp,
                  float* __restrict__ y)
{
    const int b = blockIdx.x;
    const int d = threadIdx.x;

    __shared__ float sB[DS], sC[DS];

    float An[DS], h[DS];
    #pragma unroll
    for (int n = 0; n < DS; n++) {
        An[n] = -__expf(A_log[d * DS + n]);
        h[n]  = 0.f;
    }
    const float Dv = Dp[d];

    for (int t = 0; t < TT; t++) {
        const size_t row = (size_t)(b * TT + t);
        if (d < DS)            sB[d]      = dbc[row * 48 + DTR + d];
        else if (d < 2 * DS)   sC[d - DS] = dbc[row * 48 + DTR + DS + (d - DS)];
        __syncthreads();

        const float dtv = dtb[row * DI + d];
        const float xcv = xc [row * DI + d];
        float acc = 0.f;
        #pragma unroll
        for (int n = 0; n < DS; n++) {
            const float dA = __expf(dtv * An[n]);
            h[n] = fmaf(dA, h[n], dtv * sB[n] * xcv);
            acc  = fmaf(h[n], sC[n], acc);
        }
        const float yv  = fmaf(xcv, Dv, acc);
        const float zv  = xz[row * (2 * DI) + DI + d];
        const float sig = 1.f / (1.f + __expf(-zv));
        y[row * DI + d] = yv * (zv * sig);
        __syncthreads();
    }
}

// ---------------------------------------------------------------------------
// mean over time: ybar[b][d] = mean_t y[b,t,d]   (mean commutes with out_proj)
// ---------------------------------------------------------------------------
__global__ void mean_kernel(const float* __restrict__ y, float* __restrict__ ybar)
{
    const int i = blockIdx.x * blockDim.x + threadIdx.x;
    if (i >= BB * DI) return;
    const int d = i & (DI - 1);
    const int b = i >> 9;
    const float* p = y + (size_t)(b * TT) * DI + d;
    float s = 0.f;
    for (int t = 0; t < TT; t++) s += p[(size_t)t * DI];
    ybar[i] = s * (1.f / (float)TT);
}

// ---------------------------------------------------------------------------
// head: out[b] = (ybar[b] @ out_proj_w) @ fc_w + fc_b     (32 x 10)
// ---------------------------------------------------------------------------
__global__ void head_kernel(const float* __restrict__ ybar,
                            const float* __restrict__ opw,
                            const float* __restrict__ fcw,
                            const float* __restrict__ fcb,
                            float* __restrict__ out)
{
    const int b = blockIdx.x;
    __shared__ float sy[DI];
    __shared__ float tmp[DM];
    for (int j = threadIdx.x; j < DI; j += 256) sy[j] = ybar[b * DI + j];
    __syncthreads();

    const int dm = threadIdx.x;
    float s = 0.f;
    for (int d = 0; d < DI; d++) s = fmaf(sy[d], opw[d * DM + dm], s);
    tmp[dm] = s;
    __syncthreads();

    if (threadIdx.x < 10) {
        float o = fcb[threadIdx.x];
        for (int m = 0; m < DM; m++) o = fmaf(tmp[m], fcw[m * 10 + threadIdx.x], o);
        out[b * 10 + threadIdx.x] = o;
    }
}

// ---------------------------------------------------------------------------
// launch
// ---------------------------------------------------------------------------
extern "C" void kernel_launch(void* const* d_in, const int* in_sizes, int n_in,
                              void* d_out, int out_size, void* d_ws, size_t ws_size,
                              hipStream_t stream)
{
    (void)in_sizes; (void)n_in; (void)out_size; (void)ws_size;

    const int*   x        = (const int*)  d_in[0];
    const float* emb      = (const float*)d_in[1];
    const float* enc_w1   = (const float*)d_in[2];
    const float* enc_b1   = (const float*)d_in[3];
    const float* enc_w2   = (const float*)d_in[4];
    const float* enc_b2   = (const float*)d_in[5];
    const float* conv_w   = (const float*)d_in[6];
    const float* conv_b   = (const float*)d_in[7];
    const float* in_projw = (const float*)d_in[8];
    const float* m_conv_w = (const float*)d_in[9];
    const float* m_conv_b = (const float*)d_in[10];
    const float* x_projw  = (const float*)d_in[11];
    const float* dt_projw = (const float*)d_in[12];
    const float* dt_projb = (const float*)d_in[13];
    const float* A_log    = (const float*)d_in[14];
    const float* Dp       = (const float*)d_in[15];
    const float* out_projw= (const float*)d_in[16];
    const float* fc_w     = (const float*)d_in[17];
    const float* fc_b     = (const float*)d_in[18];
    float* out = (float*)d_out;

    // workspace layout (bytes).  Region [0, 16.9MB) hosts h0/h1/h2 early and is
    // dead by scan time, so y_inner reuses it.
    char* ws = (char*)d_ws;
    float* h0    = (float*)(ws + 0);                      //  8 MB  (32 x 65536)
    float* h1    = (float*)(ws + 8388608);                // 128 KB (32 x 1024)
    float* h2    = (float*)(ws + 8519680);                //  8 MB  (32 x 65536)
    float* ybuf  = (float*)(ws + 0);                      // 16 MB  (8192 x 512) reuse
    float* wT    = (float*)(ws + 16908288);               // 640 KB
    float* u     = (float*)(ws + 17563648);               //  8 MB  (8192 x 256)
    float* xz    = (float*)(ws + 25952256);               // 32 MB  (8192 x 1024)
    float* xc    = (float*)(ws + 59506688);               // 16 MB  (8192 x 512)
    float* dbc   = (float*)(ws + 76283904);               // 1.5 MB (8192 x 48)
    float* dtb   = (float*)(ws + 77856768);               // 16 MB  (8192 x 512)
    float* ybar  = (float*)(ws + 94633984);               // 64 KB  (32 x 512)
    float* fc1p  = (float*)(ws + 94699520);               //  1 MB  (8 x 32 x 1024)

    // 1) embedding gather
    embed_kernel<<<(BB*SS*EE + 255) / 256, 256, 0, stream>>>(x, emb, h0);

    // 2) enc FC1: (32x65536)x(65536x1024), K split 8 ways across grid.z
    //    [HBM-bound: 256MB weights; 512 blocks keep all WGPs streaming]
    wmma_gemm_f32<0, false, EE*SS, HH, HH, (EE*SS)/KS_FC1, KS_FC1>
        <<<dim3(HH/16, 1, KS_FC1), 256, 0, stream>>>(h0, enc_w1, fc1p, nullptr);
    ksum_bias_relu_kernel<<<(BB*HH + 255) / 256, 256, 0, stream>>>(fc1p, enc_b1, h1);

    // 3) enc FC2: (32x1024)x(1024x65536) + b, relu   [HBM-bound: 256MB weights]
    wmma_gemm_f32<1, true, HH, EE*SS, EE*SS, HH, 1>
        <<<dim3((EE*SS)/16, 1), 256, 0, stream>>>(h1, enc_w2, h2, enc_b2);

    // 4) conv weights transpose + conv/relu/maxpool2  -> u (8192 x 256)
    convw_transpose_kernel<<<(COUT*EE*5 + 255) / 256, 256, 0, stream>>>(conv_w, wT);
    conv_pool_kernel<<<BB*TT, 256, 0, stream>>>(h2, wT, conv_b, u);

    // 5) in_proj: (8192x256)x(256x1024) -> xz
    wmma_gemm_f32<0, false, DM, 2*DI, 2*DI, DM, 1>
        <<<dim3((2*DI)/16, NTOK/32), 256, 0, stream>>>(u, in_projw, xz, nullptr);

    // 6) depthwise conv + silu -> xc
    dwconv_silu_kernel<<<(NTOK*DI + 255) / 256, 256, 0, stream>>>(
        xz, m_conv_w, m_conv_b, xc);

    // 7) x_proj: (8192x512)x(512x48) -> dbc
    wmma_gemm_f32<0, false, DI, 48, 48, DI, 1>
        <<<dim3(48/16, NTOK/32), 256, 0, stream>>>(xc, x_projw, dbc, nullptr);

    // 8) dt = softplus(dbc[:, :16] @ dt_proj_w + b)   (K = 16)
    wmma_gemm_f32<2, true, 48, DI, DI, DTR, 1>
        <<<dim3(DI/16, NTOK/32), 256, 0, stream>>>(dbc, dt_projw, dtb, dt_projb);

    // 9) selective scan (+ xc*D, * silu(z)) -> y_inner
    mamba_scan_kernel<<<BB, DI, 0, stream>>>(xc, dtb, dbc, xz, A_log, Dp, ybuf);

    // 10) mean over time (mean commutes with out_proj), then fused head
    mean_kernel<<<(BB*DI + 255) / 256, 256, 0, stream>>>(ybuf, ybar);
    head_kernel<<<BB, 256, 0, stream>>>(ybar, out_projw, fc_w, fc_b, out);
}